// Factorization_Machine_68917045232362
// MI455X (gfx1250) — compile-verified
//
#include <hip/hip_runtime.h>

// Factorization Machine forward, MI455X (gfx1250, wave32).
//
// Memory-bound gather: ~87 MB of traffic -> ~3.7 us roofline @ 23.3 TB/s.
// One wave per batch element, lane = embedding dim (K=32 == wave32).
// Each embedding row is fetched as a single fully-coalesced 128B wave load;
// indices + w1 ride the scalar (SMEM) path via readfirstlane.
// sum_sq = diag(S S^T) per 16-element block computed with 8 chained
// v_wmma_f32_16x16x4_f32 (exact f32), A==B loaded straight from LDS in the
// ISA-documented 16x4 f32 A layout.

#define NUM_FIELDS 39
#define KDIM 32

typedef __attribute__((ext_vector_type(2))) float v2f;
typedef __attribute__((ext_vector_type(8))) float v8f;

__global__ __launch_bounds__(512)
void fm_fwd_kernel(const int* __restrict__ x,      // [F, B]
                   const float* __restrict__ emb,  // [H, 32]
                   const float* __restrict__ w0p,  // [1]
                   const float* __restrict__ w1,   // [H]
                   float* __restrict__ out,        // [B]
                   int B)
{
    __shared__ float S[16][KDIM + 2];   // field-summed embeddings (+2 pad: bank spread)
    __shared__ float SQ[16];            // per-element sum of squares
    __shared__ float FM1[16];           // per-element first-order term
    __shared__ float Dlds[8 * 32];      // WMMA D-matrix bounce for diag extraction

    const int lane = threadIdx.x & 31;
    const int wave = threadIdx.x >> 5;
    const int b    = blockIdx.x * 16 + wave;   // wave-uniform batch element

    float s = 0.0f, sq = 0.0f, fm1 = 0.0f;

    if (b < B) {
        for (int f = 0; f < NUM_FIELDS; ++f) {
            // Wave-uniform index -> SGPR, so the row gather is saddr+lane*4
            // (one coalesced 128B request) and w1 is a scalar load.
            int idx = __builtin_amdgcn_readfirstlane(x[f * B + b]);
            float v = emb[(size_t)idx * KDIM + lane];
            s  += v;
            sq += v * v;
            fm1 += w1[idx];               // uniform across the wave
        }
        // Reduce sum-of-squares across the 32 lanes (per-element scalar).
        #pragma unroll
        for (int off = 16; off > 0; off >>= 1)
            sq += __shfl_xor(sq, off, 32);

        S[wave][lane] = s;
        if (lane == 0) { SQ[wave] = sq; FM1[wave] = fm1; }
    }
    __syncthreads();

    if (wave == 0) {
        // --- WMMA Gram matrix: D = S * S^T over K=32 in 8 chunks of K=4. ---
        // A (16x4 f32) layout: lane L row M=L%16; VGPR0 holds K=4c+2*(L/16),
        // VGPR1 holds K=4c+2*(L/16)+1.  B = A^T has the identical register
        // image, so one LDS load feeds both operands.
        const int r = lane & 15;
        const int h = lane >> 4;
        v8f d = {};
        #pragma unroll
        for (int c = 0; c < 8; ++c) {
            v2f a;
            a.x = S[r][4 * c + 2 * h];
            a.y = S[r][4 * c + 2 * h + 1];
            d = __builtin_amdgcn_wmma_f32_16x16x4_f32(
                    false, a, false, a, (short)0, d, false, false);
        }

        // Spill D to LDS; diag(D)[m] lives at (lane=m, VGPR m) for m<8 and
        // (lane=m+16, VGPR m-8) for m>=8 per the C/D layout tables.
        #pragma unroll
        for (int v = 0; v < 8; ++v)
            Dlds[lane * 8 + v] = d[v];

        if (lane < 16) {
            int ob = blockIdx.x * 16 + lane;
            if (ob < B) {
                float sum_sq = (lane < 8) ? Dlds[lane * 8 + lane]
                                          : Dlds[(lane + 16) * 8 + (lane - 8)];
                float logit = w0p[0] + FM1[lane] + 0.5f * (sum_sq - SQ[lane]);
                out[ob] = logit;
            }
        }
    }
}

extern "C" void kernel_launch(void* const* d_in, const int* in_sizes, int n_in,
                              void* d_out, int out_size, void* d_ws, size_t ws_size,
                              hipStream_t stream) {
    const int*   x   = (const int*)d_in[0];
    const float* emb = (const float*)d_in[1];
    const float* w0  = (const float*)d_in[2];
    const float* w1  = (const float*)d_in[3];
    float*       out = (float*)d_out;

    const int B = in_sizes[0] / NUM_FIELDS;   // 16384
    const int blocks = (B + 15) / 16;         // 16 batch elements per block
    hipLaunchKernelGGL(fm_fwd_kernel, dim3(blocks), dim3(512), 0, stream,
                       x, emb, w0, w1, out, B);
}